// PhysJointModel_6734508720622
// MI455X (gfx1250) — compile-verified
//
#include <hip/hip_runtime.h>
#include <stdint.h>

#define NJ  8      // joints
#define CT  32     // time steps per LDS chunk
#define BPW 4      // batches per wave (4 batches x 8 joints = 32 lanes)
#define WPB 2      // waves per block (independent; no barriers needed)
#define BPB (BPW * WPB)
#define CHUNK_F4 (BPW * CT * NJ)   // float4 elements per buffer per wave (1024)

// CDNA5 async copy: global memory -> LDS, tracked by ASYNCcnt.
// VDST slot = 32-bit LDS byte address VGPR, VADDR = 64-bit global address.
__device__ __forceinline__ void async_ld_b128(uint32_t lds_addr, const void* gaddr) {
    asm volatile("global_load_async_to_lds_b128 %0, %1, off"
                 :: "v"(lds_addr), "v"(gaddr)
                 : "memory");
}

__global__ __launch_bounds__(WPB * 32) void phys_joint_scan(
    const float* __restrict__ x,        // (B, T, 32)
    const float* __restrict__ state0,   // (B, 8, 2)
    const float* __restrict__ M,        // (8, 2)
    const float* __restrict__ inertia,  // (8,)
    const float* __restrict__ damping,  // (8,)
    float* __restrict__ out,            // (B, T, 8)
    float* __restrict__ muscle,         // (B, 8, 2, 2)
    float* __restrict__ finals,         // (B, 8, 2)
    int T)
{
    constexpr float DT = 1.0f / 60.0f;

    // 2 waves * 2 buffers * (4 batches * 32 steps * 8 joints) float4 = 64 KB
    __shared__ float4 lds[WPB * 2 * CHUNK_F4];

    const int lane = threadIdx.x & 31;
    const int wave = threadIdx.x >> 5;
    const int j    = lane & 7;
    const int bi   = lane >> 3;
    const int b    = blockIdx.x * BPB + wave * BPW + bi;

    // per-joint constants (lane-indexed scalar loads, done once)
    const float m0    = M[2 * j];
    const float m1    = M[2 * j + 1];
    const float m00   = m0 * m0;
    const float m11   = m1 * m1;
    const float inv_i = 1.0f / inertia[j];
    const float c     = damping[j];

    float theta  = state0[(b * NJ + j) * 2 + 0];
    float dtheta = state0[(b * NJ + j) * 2 + 1];

    // per-wave double buffer; element layout (buf, bi, tl, j) in float4 units.
    // Keep reads as direct lds[...] indexing so clang keeps addrspace(3) -> ds_load_b128.
    const int rbase0 = wave * (2 * CHUNK_F4) + bi * (CT * NJ) + j;
    const int rbase1 = rbase0 + CHUNK_F4;

    // 32-bit LDS byte addresses for the async-copy asm (low 32 bits of generic ptr)
    const uint32_t wa0 = (uint32_t)(size_t)&lds[rbase0];
    const uint32_t wa1 = (uint32_t)(size_t)&lds[rbase1];

    // per-lane global base: x + b*T*128B + j*16B ; per-step stride = 128 B
    const char* gb = (const char*)x + (size_t)b * (size_t)T * (NJ * 16) + (size_t)j * 16;
    float*      ob = out + (size_t)b * (size_t)T * NJ + j;

    const int nchunk = T / CT;

    // prefetch chunk 0 into buffer 0
#pragma unroll 4
    for (int tl = 0; tl < CT; ++tl)
        async_ld_b128(wa0 + (uint32_t)tl * (NJ * 16), gb + (size_t)tl * (NJ * 16));

    for (int cidx = 0; cidx < nchunk; ++cidx) {
        const int cur = cidx & 1;
        if (cidx + 1 < nchunk) {
            // issue next chunk into the other buffer, then wait for current chunk
            const uint32_t wnext = cur ? wa0 : wa1;
            const char*    gnext = gb + (size_t)(cidx + 1) * CT * (NJ * 16);
#pragma unroll 4
            for (int tl = 0; tl < CT; ++tl)
                async_ld_b128(wnext + (uint32_t)tl * (NJ * 16),
                              gnext + (size_t)tl * (NJ * 16));
            // loads complete in order: <=32 outstanding means chunk `cidx` is in LDS
            asm volatile("s_wait_asynccnt 0x20" ::: "memory");
        } else {
            asm volatile("s_wait_asynccnt 0x0" ::: "memory");
        }

        const int rb = cur ? rbase1 : rbase0;
        float*    o  = ob + (size_t)cidx * CT * NJ;
#pragma unroll 8
        for (int tl = 0; tl < CT; ++tl) {
            const float4 v = lds[rb + tl * NJ];      // ds_load_b128: {F0, F1, K0, K1}
            const float a   = m0 * v.x + m1 * v.y;   // M . F
            const float kk  = m00 * v.z + m11 * v.w; // M^2 . K
            const float tau = a - theta * kk - c * dtheta;
            dtheta += (tau * inv_i) * DT;
            theta  += dtheta * DT;
            o[tl * NJ] = theta;                      // out[b, t, j]
        }
    }

    // final_state (B,8,2) and muscle_states (B,8,2,2) = final_state[...,None]*M[j,:]
    const int fj = b * NJ + j;
    finals[fj * 2 + 0] = theta;
    finals[fj * 2 + 1] = dtheta;
    float4 mv;
    mv.x = theta * m0;  mv.y = theta * m1;
    mv.z = dtheta * m0; mv.w = dtheta * m1;
    *(float4*)(muscle + fj * 4) = mv;
}

extern "C" void kernel_launch(void* const* d_in, const int* in_sizes, int n_in,
                              void* d_out, int out_size, void* d_ws, size_t ws_size,
                              hipStream_t stream) {
    const float* x       = (const float*)d_in[0];
    const float* state0  = (const float*)d_in[1];
    const float* M       = (const float*)d_in[2];
    const float* inertia = (const float*)d_in[3];
    const float* damping = (const float*)d_in[4];

    const int B = in_sizes[1] / (NJ * 2);          // state0 = B*8*2
    const int T = in_sizes[0] / (B * NJ * 4);      // x = B*T*32

    float* out    = (float*)d_out;
    float* muscle = out + (size_t)B * T * NJ;      // out is (B,T,8)
    float* finals = muscle + (size_t)B * NJ * 4;   // muscle is (B,8,2,2)

    dim3 grid(B / BPB);
    dim3 block(WPB * 32);
    hipLaunchKernelGGL(phys_joint_scan, grid, block, 0, stream,
                       x, state0, M, inertia, damping, out, muscle, finals, T);
}